// GraphConvEncoder_2353642078829
// MI455X (gfx1250) — compile-verified
//
#include <hip/hip_runtime.h>
#include <hip/hip_bf16.h>

typedef __attribute__((ext_vector_type(2))) float v2f;
typedef __attribute__((ext_vector_type(8))) float v8f;

#define Gg 128
#define Hh 256
#define Ee 524288

// ---------------- GEMM: C[M x 256] = A[M x 256] * B[256 x 256], fp32 WMMA ----
// grid(M/128, 4), block 256 (8 waves). Wave w computes rows [blkX*128+16w, +16),
// cols [blkY*64, +64) as 4 independent 16x16 accumulators (breaks WMMA D->C
// RAW chains, 4x A-fragment reuse). B panel (256x64, 64KB) staged in LDS with a
// k-paired layout: B[k][n] -> Bs[(k>>1)*128 + 2n + (k&1)], so each WMMA B
// fragment {B[k][n],B[k+1][n]} is one aligned ds_load_b64 (no register repack).
__global__ __launch_bounds__(256) void gemm_wmma_f32(
    const float* __restrict__ A, const float* __restrict__ B,
    float* __restrict__ C) {
  __shared__ float Bs[256 * 64];
  const int t = threadIdx.x;
  const int lane = t & 31;
  const int wave = t >> 5;
  const int colBase = blockIdx.y << 6;
  // cooperative load of B[:, colBase:colBase+64] into LDS (k-paired layout)
#pragma unroll
  for (int r = 0; r < 64; ++r) {
    int i = t + (r << 8);            // 0..16383
    int kk = i >> 6, nn = i & 63;
    Bs[((kk >> 1) << 7) + (nn << 1) + (kk & 1)] = B[kk * 256 + colBase + nn];
  }
  __syncthreads();

  const int row0 = (blockIdx.x << 7) + (wave << 4);
  const int m = lane & 15;           // row within A-frag / col within B-frag
  const int kh = (lane >> 4) << 1;   // 0 or 2: K sub-offset per ISA layout
  const float* Arow = A + (size_t)(row0 + m) * 256 + kh;
  const float* Bp = Bs + ((lane >> 4) << 7) + (m << 1);

  v8f a0 = {}, a1 = {}, a2 = {}, a3 = {};
#pragma unroll 4
  for (int k0 = 0; k0 < 256; k0 += 4) {
    v2f a = *(const v2f*)(Arow + k0);       // A[m][k0+kh], A[m][k0+kh+1]
    const float* bq = Bp + (k0 << 5);       // k-pair row k0/4 + kh/2
    v2f b0 = *(const v2f*)(bq + 0);         // {B[kp][n], B[kp+1][n]}, tile 0
    v2f b1 = *(const v2f*)(bq + 32);        // tile 1 (n+16)
    v2f b2 = *(const v2f*)(bq + 64);        // tile 2 (n+32)
    v2f b3 = *(const v2f*)(bq + 96);        // tile 3 (n+48)
    a0 = __builtin_amdgcn_wmma_f32_16x16x4_f32(false, a, false, b0, (short)0, a0, false, false);
    a1 = __builtin_amdgcn_wmma_f32_16x16x4_f32(false, a, false, b1, (short)0, a1, false, false);
    a2 = __builtin_amdgcn_wmma_f32_16x16x4_f32(false, a, false, b2, (short)0, a2, false, false);
    a3 = __builtin_amdgcn_wmma_f32_16x16x4_f32(false, a, false, b3, (short)0, a3, false, false);
  }
  // C/D layout: VGPR r -> row r (lanes 0-15) / row r+8 (lanes 16-31)
  const int rbase = row0 + ((lane >> 4) << 3);
  float* Cp = C + (size_t)rbase * 256 + colBase + m;
#pragma unroll
  for (int r = 0; r < 8; ++r) {
    Cp[(size_t)r * 256 + 0]  = a0[r];
    Cp[(size_t)r * 256 + 16] = a1[r];
    Cp[(size_t)r * 256 + 32] = a2[r];
    Cp[(size_t)r * 256 + 48] = a3[r];
  }
}

// ---------------- small utility kernels ----------------
__global__ void fill_f32(float* __restrict__ p, float v, int n) {
  int i = blockIdx.x * 256 + threadIdx.x;
  if (i < n) p[i] = v;
}

__global__ void deg_edges(const int* __restrict__ dst,
                          const float* __restrict__ emask,
                          float* __restrict__ deg) {
  int e = blockIdx.x * 256 + threadIdx.x;
  if (e < Ee) {
    float m = emask[e];
    if (m != 0.0f) atomicAdd(&deg[dst[e]], m);
  }
}

__global__ void rsqrt_ip(float* __restrict__ deg, int n) {
  int i = blockIdx.x * 256 + threadIdx.x;
  if (i < n) deg[i] = rsqrtf(deg[i]);
}

// agg = h * dis^2 (self-loop message). block = node, thread = feature
__global__ void selfloop_init(const float* __restrict__ h,
                              const float* __restrict__ dis,
                              float* __restrict__ agg) {
  int i = blockIdx.x;
  int j = threadIdx.x;
  float d = dis[i];
  agg[(size_t)i * Hh + j] = h[(size_t)i * Hh + j] * d * d;
}

// one block per edge: agg[dst] += h[src] * dis[src]*dis[dst]*emask
__global__ void edge_agg(const int* __restrict__ src, const int* __restrict__ dst,
                         const float* __restrict__ emask,
                         const float* __restrict__ dis,
                         const float* __restrict__ h, float* __restrict__ agg) {
  __shared__ int ss, sd;
  __shared__ float sc;
  int e = blockIdx.x, t = threadIdx.x;
  if (t == 0) {
    int s = src[e], d = dst[e];
    ss = s; sd = d;
    sc = dis[s] * dis[d] * emask[e];
  }
  __syncthreads();
  float c = sc;
  if (c != 0.0f)
    atomicAdd(&agg[(size_t)sd * Hh + t], h[(size_t)ss * Hh + t] * c);
}

__global__ void bias_relu(float* __restrict__ agg, const float* __restrict__ b) {
  int i = blockIdx.x, j = threadIdx.x;
  float v = agg[(size_t)i * Hh + j] + b[j];
  agg[(size_t)i * Hh + j] = v > 0.0f ? v : 0.0f;
}

__global__ void invnorm256(const float* __restrict__ p, float* __restrict__ o) {
  __shared__ float red[256];
  int t = threadIdx.x;
  float v = p[t];
  red[t] = v * v;
  __syncthreads();
  for (int s = 128; s > 0; s >>= 1) {
    if (t < s) red[t] += red[t + s];
    __syncthreads();
  }
  if (t == 0) o[0] = rsqrtf(red[0]);
}

// one wave per row: out[i] = f(dot(x[i,:], v)); f = tanh(scale*dot) or dot+bias
__global__ __launch_bounds__(256) void rowdot(
    const float* __restrict__ x, const float* __restrict__ v,
    const float* __restrict__ scale, const float* __restrict__ bias,
    float* __restrict__ out, int M, int useTanh) {
  int w = (blockIdx.x * 256 + threadIdx.x) >> 5;
  int lane = threadIdx.x & 31;
  if (w >= M) return;
  const float* row = x + (size_t)w * Hh;
  float acc = 0.0f;
#pragma unroll
  for (int j = 0; j < 8; ++j) {
    int c = lane + (j << 5);
    acc += row[c] * v[c];
  }
#pragma unroll
  for (int off = 16; off > 0; off >>= 1) acc += __shfl_xor(acc, off, 32);
  if (lane == 0) {
    float r = acc;
    if (scale) r *= scale[0];
    if (bias) r += bias[0];
    out[w] = useTanh ? tanhf(r) : r;
  }
}

// block per graph: bitonic top-k (descending) over n<=512 scores
__global__ __launch_bounds__(256) void topk_pool(
    const float* __restrict__ score, int n, int k,
    int* __restrict__ keep, int* __restrict__ mapping) {
  __shared__ float s[512];
  __shared__ int id[512];
  int g = blockIdx.x, t = threadIdx.x;
  for (int i = t; i < 512; i += 256) {
    s[i] = (i < n) ? score[g * n + i] : -3.0e38f;
    id[i] = i;
  }
  for (int i = t; i < n; i += 256) mapping[g * n + i] = -1;
  __syncthreads();
  for (int size = 2; size <= 512; size <<= 1) {
    for (int stride = size >> 1; stride > 0; stride >>= 1) {
      int pos = 2 * t - (t & (stride - 1));
      bool down = ((pos & size) == 0);  // descending overall
      float a = s[pos], b = s[pos + stride];
      bool sw = down ? (a < b) : (a > b);
      if (sw) {
        s[pos] = b; s[pos + stride] = a;
        int tmp = id[pos]; id[pos] = id[pos + stride]; id[pos + stride] = tmp;
      }
      __syncthreads();
    }
  }
  for (int r = t; r < k; r += 256) {
    int old = id[r];
    keep[g * k + r] = g * n + old;
    mapping[g * n + old] = g * k + r;
  }
}

// block per new node: xnew[m] = hact[keep[m]] * score[keep[m]]
__global__ void gather_scale(const float* __restrict__ hact,
                             const int* __restrict__ keep,
                             const float* __restrict__ score,
                             float* __restrict__ xnew) {
  __shared__ int so;
  __shared__ float ssc;
  int m = blockIdx.x, t = threadIdx.x;
  if (t == 0) { so = keep[m]; ssc = score[so]; }
  __syncthreads();
  xnew[(size_t)m * Hh + t] = hact[(size_t)so * Hh + t] * ssc;
}

__global__ void edge_remap(const int* __restrict__ srcI, const int* __restrict__ dstI,
                           const float* __restrict__ emI,
                           const int* __restrict__ mapping,
                           int* __restrict__ srcO, int* __restrict__ dstO,
                           float* __restrict__ emO) {
  int e = blockIdx.x * 256 + threadIdx.x;
  if (e >= Ee) return;
  int ns = mapping[srcI[e]];
  int nd = mapping[dstI[e]];
  bool valid = (ns >= 0) && (nd >= 0) && (emI[e] > 0.0f);
  srcO[e] = valid ? ns : 0;
  dstO[e] = valid ? nd : 0;
  emO[e] = valid ? 1.0f : 0.0f;
}

// block per graph: per-graph softmax over k gates, weighted sum into out (+=)
__global__ __launch_bounds__(256) void att_pool(
    const float* __restrict__ x, const float* __restrict__ gate,
    float* __restrict__ out, int k) {
  __shared__ float red[256];
  __shared__ float w[512];
  int g = blockIdx.x, t = threadIdx.x;
  const float* gg = gate + (size_t)g * k;
  float mx = -3.0e38f;
  for (int r = t; r < k; r += 256) mx = fmaxf(mx, gg[r]);
  red[t] = mx;
  __syncthreads();
  for (int s = 128; s > 0; s >>= 1) {
    if (t < s) red[t] = fmaxf(red[t], red[t + s]);
    __syncthreads();
  }
  mx = red[0];
  __syncthreads();
  float sum = 0.0f;
  for (int r = t; r < k; r += 256) {
    float e = expf(gg[r] - mx);
    w[r] = e;
    sum += e;
  }
  red[t] = sum;
  __syncthreads();
  for (int s = 128; s > 0; s >>= 1) {
    if (t < s) red[t] += red[t + s];
    __syncthreads();
  }
  float invZ = 1.0f / red[0];
  __syncthreads();
  float acc = 0.0f;
  const float* xg = x + (size_t)g * k * Hh;
  for (int r = 0; r < k; ++r) acc += w[r] * invZ * xg[(size_t)r * Hh + t];
  out[(size_t)g * Hh + t] += acc;
}

extern "C" void kernel_launch(void* const* d_in, const int* in_sizes, int n_in,
                              void* d_out, int out_size, void* d_ws, size_t ws_size,
                              hipStream_t stream) {
  const float* x = (const float*)d_in[0];
  const int* es = (const int*)d_in[1];
  const int* ed = (const int*)d_in[2];
  const float* W[3] = {(const float*)d_in[3], (const float*)d_in[5], (const float*)d_in[7]};
  const float* Bv[3] = {(const float*)d_in[4], (const float*)d_in[6], (const float*)d_in[8]};
  const float* P[3] = {(const float*)d_in[9], (const float*)d_in[10], (const float*)d_in[11]};
  const float* attw = (const float*)d_in[12];
  const float* attb = (const float*)d_in[13];
  float* out = (float*)d_out;

  char* base = (char*)d_ws;
  size_t off = 0;
  auto alloc = [&](size_t bytes) -> void* {
    void* p = base + off;
    off = (off + bytes + 255) & ~(size_t)255;
    return p;
  };
  float* hbuf = (float*)alloc(65536ull * Hh * 4);   // GEMM output
  float* agg  = (float*)alloc(65536ull * Hh * 4);   // aggregate / activated
  float* xnew = (float*)alloc(52480ull * Hh * 4);   // pooled features
  float* deg  = (float*)alloc(65536ull * 4);        // degree -> dis (in place)
  float* scr  = (float*)alloc(65536ull * 4);        // node scores
  float* gat  = (float*)alloc(52480ull * 4);        // attention gates
  float* invn = (float*)alloc(256);
  int* mapping = (int*)alloc(65536ull * 4);
  int* keep    = (int*)alloc(52480ull * 4);
  int* srcW    = (int*)alloc((size_t)Ee * 4);
  int* dstW    = (int*)alloc((size_t)Ee * 4);
  float* emk   = (float*)alloc((size_t)Ee * 4);
  (void)ws_size; (void)in_sizes; (void)n_in; (void)out_size;

  const int nL[4] = {512, 410, 328, 263};  // ceil(0.8*n) cascade

  fill_f32<<<(Ee + 255) / 256, 256, 0, stream>>>(emk, 1.0f, Ee);
  fill_f32<<<(Gg * Hh + 255) / 256, 256, 0, stream>>>(out, 0.0f, Gg * Hh);

  const int* curSrc = es;
  const int* curDst = ed;
  const float* xin = x;

  for (int L = 0; L < 3; ++L) {
    const int Min = Gg * nL[L];
    const int k = nL[L + 1];
    const int Mout = Gg * k;

    // 1) h = xin @ W (WMMA fp32, 128x64 block tiles)
    gemm_wmma_f32<<<dim3(Min / 128, 4), 256, 0, stream>>>(xin, W[L], hbuf);
    // 2) deg = 1 + segment_sum(emask, dst); dis = rsqrt(deg)
    fill_f32<<<(Min + 255) / 256, 256, 0, stream>>>(deg, 1.0f, Min);
    deg_edges<<<Ee / 256, 256, 0, stream>>>(curDst, emk, deg);
    rsqrt_ip<<<(Min + 255) / 256, 256, 0, stream>>>(deg, Min);
    // 3) agg = h*dis^2 + sum_e coef*h[src]; relu(agg + b)
    selfloop_init<<<Min, 256, 0, stream>>>(hbuf, deg, agg);
    edge_agg<<<Ee, 256, 0, stream>>>(curSrc, curDst, emk, deg, hbuf, agg);
    bias_relu<<<Min, 256, 0, stream>>>(agg, Bv[L]);
    // 4) score = tanh((h.p)/|p|)
    invnorm256<<<1, 256, 0, stream>>>(P[L], invn);
    rowdot<<<Min / 8, 256, 0, stream>>>(agg, P[L], invn, nullptr, scr, Min, 1);
    // 5) per-graph top-k, gather + scale, edge remap
    topk_pool<<<Gg, 256, 0, stream>>>(scr, nL[L], k, keep, mapping);
    gather_scale<<<Mout, 256, 0, stream>>>(agg, keep, scr, xnew);
    edge_remap<<<Ee / 256, 256, 0, stream>>>(curSrc, curDst, emk, mapping,
                                             srcW, dstW, emk);
    // 6) global attention pool, accumulate into out
    rowdot<<<Mout / 8, 256, 0, stream>>>(xnew, attw, nullptr, attb, gat, Mout, 0);
    att_pool<<<Gg, 256, 0, stream>>>(xnew, gat, out, k);

    curSrc = srcW;
    curDst = dstW;
    xin = xnew;
  }
}